// Model_39676907885045
// MI455X (gfx1250) — compile-verified
//
#include <hip/hip_runtime.h>
#include <stdint.h>

typedef __attribute__((ext_vector_type(2))) float v2f;
typedef __attribute__((ext_vector_type(8))) float v8f;

#define BATCH 16
#define LQ    2048
#define LK    2048
#define DH    128

#define ROWS_PER_BLOCK 16
#define WAVES          8
#define TILES_PER_WAVE 16   // 16 tiles * 16 cols = 256 cols/wave; 8 waves cover 2048

__device__ __forceinline__ uint32_t rotl32(uint32_t x, int r) {
  return (x << r) | (x >> (32 - r));
}

// Exact JAX threefry2x32 for key = jax.random.key(42) -> key data (0, 42).
// random_bits over a 2^26 iota pairs counters (i, i + 2^25); element i < 2^25
// takes lane-0 output, otherwise lane-1 output.
__device__ __forceinline__ uint32_t jax_bits(uint32_t idx) {
  const uint32_t HALF = 1u << 25;
  const uint32_t K0 = 0u, K1 = 42u;
  const uint32_t K2 = K0 ^ K1 ^ 0x1BD11BDAu;
  const uint32_t ks[3] = {K0, K1, K2};
  const int rot[2][4] = {{13, 15, 26, 6}, {17, 29, 16, 24}};
  bool second = idx >= HALF;
  uint32_t x0 = second ? (idx - HALF) : idx;
  uint32_t x1 = second ? idx : (idx + HALF);
  x0 += K0;
  x1 += K1;
#pragma unroll
  for (int i = 0; i < 5; ++i) {
#pragma unroll
    for (int jr = 0; jr < 4; ++jr) {
      x0 += x1;
      x1 = rotl32(x1, rot[i & 1][jr]);
      x1 ^= x0;
    }
    x0 += ks[(i + 1) % 3];
    x1 += ks[(i + 2) % 3] + (uint32_t)(i + 1);
  }
  return second ? x1 : x0;
}

__global__ __launch_bounds__(256) void attn_probs_dropout(
    const float* __restrict__ Q, const float* __restrict__ K,
    const float* __restrict__ X1, float* __restrict__ Out) {
  const int blk   = blockIdx.x;
  const int b     = blk >> 7;    // / (LQ/16)
  const int qt    = blk & 127;
  const int qbase = qt * ROWS_PER_BLOCK;
  const int wave  = threadIdx.x >> 5;
  const int lane  = threadIdx.x & 31;
  const int l16   = lane & 15;
  const int half  = lane >> 4;
  const int colbase = wave * (TILES_PER_WAVE * 16);

  __shared__ float red[WAVES * ROWS_PER_BLOCK];

  // ---- A fragments: Q rows qbase..qbase+15, D = 0..127 ----
  // f32 16x4 A layout: lanes 0-15 row=l16 K={4k,4k+1}; lanes 16-31 K={4k+2,4k+3}
  const float* qptr = Q + ((size_t)b * LQ + qbase) * DH;
  v2f a[32];
#pragma unroll
  for (int k = 0; k < 32; ++k)
    a[k] = *(const v2f*)(qptr + (size_t)l16 * DH + 4 * k + 2 * half);

  // ---- GEMM: per wave 16 tiles of 16x16, K=128 via 32x v_wmma_f32_16x16x4_f32
  // B[d][n] = Key[n][d]; B fragment layout mirrors A (lane = column n).
  const float* kptr = K + (size_t)b * LK * DH;
  v8f c[TILES_PER_WAVE];
#pragma unroll
  for (int j = 0; j < TILES_PER_WAVE; ++j) {
    const int nb = colbase + j * 16;
    const float* kt = kptr + (size_t)(nb + l16) * DH + 2 * half;
    v8f acc = {};
#pragma unroll
    for (int k = 0; k < 32; ++k) {
      v2f bf = *(const v2f*)(kt + 4 * k);
      acc = __builtin_amdgcn_wmma_f32_16x16x4_f32(false, a[k], false, bf,
                                                  (short)0, acc, false, false);
    }
    c[j] = acc;
  }

  // ---- scale by x1 (broadcast over batch) and running row max ----
  // C layout: vgpr r -> row (r + 8*half), col l16 within tile
  float rmax[8];
#pragma unroll
  for (int r = 0; r < 8; ++r) rmax[r] = -3.402823466e38f;
#pragma unroll
  for (int j = 0; j < TILES_PER_WAVE; ++j) {
    const int col = colbase + j * 16 + l16;
#pragma unroll
    for (int r = 0; r < 8; ++r) {
      const int row = r + 8 * half;
      float xs = X1[(size_t)(qbase + row) * LK + col];
      float v = c[j][r] * xs;
      c[j][r] = v;
      rmax[r] = fmaxf(rmax[r], v);
    }
  }
  // reduce across the 16 lanes of each half-wave (xor masks stay inside half)
#pragma unroll
  for (int r = 0; r < 8; ++r) {
#pragma unroll
    for (int m = 1; m < 16; m <<= 1)
      rmax[r] = fmaxf(rmax[r], __shfl_xor(rmax[r], m, 32));
  }
  // cross-wave reduction through LDS
  if (l16 == 0) {
#pragma unroll
    for (int r = 0; r < 8; ++r) red[wave * 16 + 8 * half + r] = rmax[r];
  }
  __syncthreads();
  float fm[8];
#pragma unroll
  for (int r = 0; r < 8; ++r) {
    const int row = r + 8 * half;
    float v = red[row];
    for (int w2 = 1; w2 < WAVES; ++w2) v = fmaxf(v, red[w2 * 16 + row]);
    fm[r] = v;
  }
  __syncthreads();  // red is reused for sums below

  // ---- exp(x - max) in place, row sums ----
  float rsum[8];
#pragma unroll
  for (int r = 0; r < 8; ++r) rsum[r] = 0.0f;
#pragma unroll
  for (int j = 0; j < TILES_PER_WAVE; ++j) {
#pragma unroll
    for (int r = 0; r < 8; ++r) {
      float e = __expf(c[j][r] - fm[r]);
      c[j][r] = e;
      rsum[r] += e;
    }
  }
#pragma unroll
  for (int r = 0; r < 8; ++r) {
#pragma unroll
    for (int m = 1; m < 16; m <<= 1) rsum[r] += __shfl_xor(rsum[r], m, 32);
  }
  if (l16 == 0) {
#pragma unroll
    for (int r = 0; r < 8; ++r) red[wave * 16 + 8 * half + r] = rsum[r];
  }
  __syncthreads();
  float inv[8];
#pragma unroll
  for (int r = 0; r < 8; ++r) {
    const int row = r + 8 * half;
    float s = 0.0f;
    for (int w2 = 0; w2 < WAVES; ++w2) s += red[w2 * 16 + row];
    inv[r] = 2.0f / s;  // softmax normalize * 1/(1-p), p = 0.5
  }

  // ---- fused dropout (exact threefry mask) + store ----
  const size_t obase = ((size_t)b * LQ + qbase) * (size_t)LK;
#pragma unroll
  for (int j = 0; j < TILES_PER_WAVE; ++j) {
    const int col = colbase + j * 16 + l16;
#pragma unroll
    for (int r = 0; r < 8; ++r) {
      const int row = r + 8 * half;
      const size_t idx = obase + (size_t)row * LK + col;  // < 2^26, fits u32
      uint32_t bits = jax_bits((uint32_t)idx);
      // keep iff uniform < 0.5  <=>  bit31 == 0
      Out[idx] = (bits >> 31) ? 0.0f : c[j][r] * inv[r];
    }
  }
}

extern "C" void kernel_launch(void* const* d_in, const int* in_sizes, int n_in,
                              void* d_out, int out_size, void* d_ws, size_t ws_size,
                              hipStream_t stream) {
  const float* Q  = (const float*)d_in[0];
  const float* K  = (const float*)d_in[1];
  // d_in[2] (value) is dead code in the reference
  const float* X1 = (const float*)d_in[3];
  float* Out = (float*)d_out;

  dim3 grid(BATCH * (LQ / ROWS_PER_BLOCK));  // 16 * 128 = 2048 blocks
  dim3 block(256);                           // 8 waves (wave32)
  hipLaunchKernelGGL(attn_probs_dropout, grid, block, 0, stream, Q, K, X1, Out);
}